// DMC_58299886076023
// MI455X (gfx1250) — compile-verified
//
#include <hip/hip_runtime.h>
#include <stdint.h>

// ---------------------------------------------------------------------------
// Dual marching cubes, GRID=192, ISO=0, fp32.
// Outputs (concatenated flat in d_out as float32):
//   verts : NC*3   (NC = 193^3 = 7,189,057)
//   vmask : NC
//   quads : NE3*4  (NE3 = 3*193*192*192 = 21,344,256)   (int ids as float)
//   qmask : NE3
// Store-bound: ~542 MB of output vs 28 MB input (L2-resident).
// All output stores are non-temporal (TH_NT) so the write stream does not
// evict the heavily reused input grid from L2/WGP$.
// ---------------------------------------------------------------------------

#define G      192
#define C      193              // cells per axis (G+1)
#define NC     (193u*193u*193u) // 7,189,057
#define EPERD  (193u*192u*192u) // 7,114,752 edges per direction
#define NE3    (3u*EPERD)       // 21,344,256

// tile of cells per workgroup
#define TX 8
#define TY 8
#define TZ 32
// halo'd grid region per tile: (TX+1) x (TY+1) x (TZ+1)
#define LDS_ELEMS ((TX+1)*(TY+1)*(TZ+1))   // 9*9*33 = 2673 floats

typedef unsigned int u32x4 __attribute__((ext_vector_type(4)));
typedef int          i32x8 __attribute__((ext_vector_type(8)));
typedef int          i32x4 __attribute__((ext_vector_type(4)));
typedef float        f32x4 __attribute__((ext_vector_type(4)));  // 16B, NT-storable

// one cube edge: A/B corner indices (idx = dx + 2*dy + 4*dz), A-corner offset
// (OX,OY,OZ), axis AX carrying the interpolation parameter t.
#define DO_EDGE(IA, IB, OX, OY, OZ, AX)                                   \
  {                                                                       \
    float vA = v[IA], vB = v[IB];                                         \
    if ((vA < 0.f) != (vB < 0.f)) {                                       \
      float t = (-vA) * __builtin_amdgcn_rcpf(vB - vA);                   \
      sx += (float)(OX) + ((AX) == 0 ? t : 0.f);                          \
      sy += (float)(OY) + ((AX) == 1 ? t : 0.f);                          \
      sz += (float)(OZ) + ((AX) == 2 ? t : 0.f);                          \
      cnt += 1.f;                                                         \
    }                                                                     \
  }

__global__ __launch_bounds__(256)
void dmc_verts_kernel(const float* __restrict__ grid,
                      float* __restrict__ verts,
                      float* __restrict__ vmask) {
  __shared__ float tile[LDS_ELEMS];

  // cell-tile origin: z fastest for coalescing
  const int z0 = (int)blockIdx.x * TZ;
  const int y0 = (int)blockIdx.y * TY;
  const int x0 = (int)blockIdx.z * TX;

  // clamped grid region this tile needs: [x0-1 .. x0+TX-1] etc., within [0,191]
  const int xlo = max(x0 - 1, 0), xhi = min(x0 + TX - 1, G - 1);
  const int ylo = max(y0 - 1, 0), yhi = min(y0 + TY - 1, G - 1);
  const int zlo = max(z0 - 1, 0), zhi = min(z0 + TZ - 1, G - 1);
  const int dimX = xhi - xlo + 1;   // <= TX+1
  const int dimY = yhi - ylo + 1;   // <= TY+1
  const int dimZ = zhi - zlo + 1;   // <= TZ+1

  // --- Tensor Data Mover: DMA the (dimX x dimY x dimZ) region into LDS ------
  // D# layout per CDNA5 ISA ch.8 (CDNA5 ISA).
  // dim0 = z (contiguous), dim1 = y (stride 192), dim2 = x (stride 192*192).
  {
    const uint64_t gaddr =
        (uint64_t)(const void*)grid +
        4ull * ((uint64_t)zlo + (uint64_t)G * ((uint64_t)ylo + (uint64_t)G * (uint64_t)xlo));
    const uint32_t lds_off = (uint32_t)(uintptr_t)(void*)&tile[0];

    u32x4 g0;
    g0.x = 1u;                                   // count=1, is_restore=0, gather off
    g0.y = lds_off;                              // lds_addr (bytes)
    g0.z = (uint32_t)gaddr;                      // global_addr[31:0]
    g0.w = (uint32_t)((gaddr >> 32) & 0x01FFFFFFu) | (2u << 30); // addr[56:32] | type=2

    const uint32_t tdim0 = (uint32_t)(G - zlo);  // remaining z (never hit: clamped)
    const uint32_t tdim1 = (uint32_t)(G - ylo);
    const uint32_t str0  = (uint32_t)G;          // elements between y-lines
    const uint32_t str1  = (uint32_t)(G * G);    // elements between x-slices

    i32x8 g1;
    g1[0] = (int)(2u << 16);                     // wg_mask=0, data_size=2 (4B), no flags
    g1[1] = (int)((tdim0 & 0xFFFFu) << 16);      // atomic_bar_addr=0 | tensor_dim0 lo16
    g1[2] = (int)((tdim0 >> 16) | ((tdim1 & 0xFFFFu) << 16));
    g1[3] = (int)((tdim1 >> 16) | ((uint32_t)dimZ << 16));   // tile_dim0 = z extent
    g1[4] = (int)((uint32_t)dimY | ((uint32_t)dimX << 16));  // tile_dim1, tile_dim2
    g1[5] = (int)str0;                           // tensor_dim0_stride[31:0]
    g1[6] = (int)((str1 & 0xFFFFu) << 16);       // stride0 hi16=0 | stride1 lo16
    g1[7] = (int)(str1 >> 16);                   // stride1 hi

    i32x4 g2;
    g2[0] = (int)(uint32_t)(G - xlo);            // tensor_dim2
    g2[1] = 0;                                   // tensor_dim3 (unused)
    g2[2] = 0;                                   // tensor_dim2_stride lo (unused)
    g2[3] = 0;                                   // stride hi | tile_dim3=0
    i32x4 g3 = {0, 0, 0, 0};                     // tensor_dim3_stride etc.
    i32x8 g4 = {0, 0, 0, 0, 0, 0, 0, 0};         // extra group (clang-23 6-arg form)

    if (threadIdx.x < 32u) {                     // one wave issues the DMA
      __builtin_amdgcn_tensor_load_to_lds(g0, g1, g2, g3, g4, 0);
      __builtin_amdgcn_s_wait_tensorcnt(0);
    }
  }
  __syncthreads();

  // --- per-cell dual vertex ------------------------------------------------
  const int tz = (int)(threadIdx.x & 31u);
  const int ty = (int)(threadIdx.x >> 5);        // 0..7
  const int c  = z0 + tz;
  const int b  = y0 + ty;
  const bool okbc = (c < C) && (b < C);
  const float scale = 1.0f / (float)(G - 1);     // 1/191

#pragma unroll
  for (int txl = 0; txl < TX; ++txl) {
    const int a = x0 + txl;
    if (!okbc || a >= C) continue;

    float v[8];
#pragma unroll
    for (int ci = 0; ci < 8; ++ci) {
      const int gx = a + (ci & 1) - 1;
      const int gy = b + ((ci >> 1) & 1) - 1;
      const int gz = c + ((ci >> 2) & 1) - 1;
      const bool in = ((unsigned)gx < (unsigned)G) &
                      ((unsigned)gy < (unsigned)G) &
                      ((unsigned)gz < (unsigned)G);
      v[ci] = in ? tile[((gx - xlo) * dimY + (gy - ylo)) * dimZ + (gz - zlo)]
                 : 1.0f;                          // padded value iso+1.0
    }

    float sx = 0.f, sy = 0.f, sz = 0.f, cnt = 0.f;
    DO_EDGE(0, 1, 0, 0, 0, 0)   // x-directed
    DO_EDGE(2, 3, 0, 1, 0, 0)
    DO_EDGE(4, 5, 0, 0, 1, 0)
    DO_EDGE(6, 7, 0, 1, 1, 0)
    DO_EDGE(0, 2, 0, 0, 0, 1)   // y-directed
    DO_EDGE(1, 3, 1, 0, 0, 1)
    DO_EDGE(4, 6, 0, 0, 1, 1)
    DO_EDGE(5, 7, 1, 0, 1, 1)
    DO_EDGE(0, 4, 0, 0, 0, 2)   // z-directed
    DO_EDGE(1, 5, 1, 0, 0, 2)
    DO_EDGE(2, 6, 0, 1, 0, 2)
    DO_EDGE(3, 7, 1, 1, 0, 2)

    const float inv = __builtin_amdgcn_rcpf(fmaxf(cnt, 1.0f));
    const size_t idx = ((size_t)a * C + (size_t)b) * C + (size_t)c;
    // streaming (non-temporal) stores: do not pollute L2 with write-once data
    __builtin_nontemporal_store(((float)a + sx * inv - 1.0f) * scale, &verts[idx * 3 + 0]);
    __builtin_nontemporal_store(((float)b + sy * inv - 1.0f) * scale, &verts[idx * 3 + 1]);
    __builtin_nontemporal_store(((float)c + sz * inv - 1.0f) * scale, &verts[idx * 3 + 2]);
    __builtin_nontemporal_store((cnt > 0.f) ? 1.0f : 0.0f, &vmask[idx]);
  }
}

// sign of padded grid at P(a,b,c): grid[a-1,b-1,c-1] < 0, OOB -> (1.0<0)=false
__device__ __forceinline__ bool sgnP(const float* __restrict__ g,
                                     int pa, int pb, int pc) {
  const int gx = pa - 1, gy = pb - 1, gz = pc - 1;
  if (((unsigned)gx >= (unsigned)G) | ((unsigned)gy >= (unsigned)G) |
      ((unsigned)gz >= (unsigned)G))
    return false;
  return g[((size_t)gx * G + (size_t)gy) * G + (size_t)gz] < 0.f;
}

__device__ __forceinline__ float cidf(int a, int b, int c) {
  // flat cell id; < 193^3 = 7.19M, exactly representable in fp32 (< 2^24)
  return (float)((a * C + b) * C + c);
}

__global__ __launch_bounds__(256)
void dmc_quads_kernel(const float* __restrict__ grid,
                      f32x4* __restrict__ quads,
                      float* __restrict__ qmask) {
  const uint32_t n = blockIdx.x * 256u + threadIdx.x;
  if (n >= NE3) return;

  int a, b, c, d;
  if (n < EPERD) {                 // x-directed edges, shape (193,192,192)
    d = 0;
    const uint32_t r  = n;
    const uint32_t i0 = r / (192u * 192u);
    const uint32_t r2 = r - i0 * (192u * 192u);
    const uint32_t i1 = r2 / 192u;
    const uint32_t i2 = r2 - i1 * 192u;
    a = (int)i0; b = 1 + (int)i1; c = 1 + (int)i2;
  } else if (n < 2u * EPERD) {     // y-directed edges, shape (192,193,192)
    d = 1;
    const uint32_t r  = n - EPERD;
    const uint32_t i0 = r / (193u * 192u);
    const uint32_t r2 = r - i0 * (193u * 192u);
    const uint32_t i1 = r2 / 192u;
    const uint32_t i2 = r2 - i1 * 192u;
    a = 1 + (int)i0; b = (int)i1; c = 1 + (int)i2;
  } else {                         // z-directed edges, shape (192,192,193)
    d = 2;
    const uint32_t r  = n - 2u * EPERD;
    const uint32_t i0 = r / (192u * 193u);
    const uint32_t r2 = r - i0 * (192u * 193u);
    const uint32_t i1 = r2 / 193u;
    const uint32_t i2 = r2 - i1 * 193u;
    a = 1 + (int)i0; b = 1 + (int)i1; c = (int)i2;
  }

  bool m;
  f32x4 q;
  if (d == 0) {
    m = sgnP(grid, a, b, c) != sgnP(grid, a + 1, b, c);
    q = (f32x4){cidf(a, b - 1, c - 1), cidf(a, b, c - 1),
                cidf(a, b, c),         cidf(a, b - 1, c)};
  } else if (d == 1) {
    m = sgnP(grid, a, b, c) != sgnP(grid, a, b + 1, c);
    q = (f32x4){cidf(a - 1, b, c - 1), cidf(a, b, c - 1),
                cidf(a, b, c),         cidf(a - 1, b, c)};
  } else {
    m = sgnP(grid, a, b, c) != sgnP(grid, a, b, c + 1);
    q = (f32x4){cidf(a - 1, b - 1, c), cidf(a, b - 1, c),
                cidf(a, b, c),         cidf(a - 1, b, c)};
  }

  // 16B-aligned streaming store -> global_store_b128 (non-temporal)
  __builtin_nontemporal_store(q, &quads[n]);
  __builtin_nontemporal_store(m ? 1.0f : 0.0f, &qmask[n]);
}

extern "C" void kernel_launch(void* const* d_in, const int* in_sizes, int n_in,
                              void* d_out, int out_size, void* d_ws, size_t ws_size,
                              hipStream_t stream) {
  (void)in_sizes; (void)n_in; (void)d_ws; (void)ws_size; (void)out_size;
  const float* grid = (const float*)d_in[0];
  float* out = (float*)d_out;

  float* verts = out;                          // NC*3
  float* vmask = out + (size_t)3 * NC;         // NC
  float* quads = out + (size_t)4 * NC;         // NE3*4 (16B-aligned: 4*NC % 4 == 0)
  float* qmask = out + (size_t)4 * NC + (size_t)4 * NE3;

  // verts/vmask: one block per 8x8x32 cell tile; 193 cells/axis
  dim3 g1((C + TZ - 1) / TZ,   // 7  z-tiles (fastest -> coalesced stores)
          (C + TY - 1) / TY,   // 25 y-tiles
          (C + TX - 1) / TX);  // 25 x-tiles
  dmc_verts_kernel<<<g1, 256, 0, stream>>>(grid, verts, vmask);

  // quads/qmask: one thread per edge
  const uint32_t nblk = (NE3 + 255u) / 256u;   // 83,376
  dmc_quads_kernel<<<nblk, 256, 0, stream>>>(grid, (f32x4*)quads, qmask);
}